// HFGptOssTopKRouter_82197084111418
// MI455X (gfx1250) — compile-verified
//
#include <hip/hip_runtime.h>
#include <hip/hip_bf16.h>

typedef __attribute__((ext_vector_type(2))) float v2f;
typedef __attribute__((ext_vector_type(4))) float v4f;
typedef __attribute__((ext_vector_type(8))) float v8f;
typedef __attribute__((ext_vector_type(4))) int   i4;

#define LDS_STRIDE 132   // logits staging row stride (pad vs 64 banks)
#define KC 32            // K-chunk staged per double-buffer step
#define BSTRIDE 36       // dwords per staged weight row: 16B-aligned, conflict-free

// Address-space-qualified int4 pointee types for the async-LDS builtin:
// param 1: AS1 (global) i4*, param 2: AS3 (LDS) i4*.
typedef __attribute__((address_space(1))) i4 g_i4;
typedef __attribute__((address_space(3))) i4 l_i4;

#if __has_builtin(__builtin_amdgcn_global_load_async_to_lds_b128)
#define HAVE_ASYNC_LDS 1
#else
#define HAVE_ASYNC_LDS 0
#endif

__device__ __forceinline__ void wait_asynccnt_8() {
#if __has_builtin(__builtin_amdgcn_s_wait_asynccnt)
    __builtin_amdgcn_s_wait_asynccnt(8);
#elif HAVE_ASYNC_LDS
    asm volatile("s_wait_asynccnt 8" ::: "memory");
#endif
}
__device__ __forceinline__ void wait_asynccnt_0() {
#if __has_builtin(__builtin_amdgcn_s_wait_asynccnt)
    __builtin_amdgcn_s_wait_asynccnt(0);
#elif HAVE_ASYNC_LDS
    asm volatile("s_wait_asynccnt 0" ::: "memory");
#endif
}

// Cooperatively stage W[0:128][kc:kc+KC] into dstBuf (async to LDS if available).
// 128 rows x 128B/row = 1024 b128 transfers, 8 per thread (128 threads).
__device__ __forceinline__ void stage_weights(const float* __restrict__ Wt, int H,
                                              int kc, float* dstBuf, int tid)
{
    #pragma unroll
    for (int j = 0; j < 8; ++j) {
        const int id  = j * 128 + tid;
        const int row = id >> 3;      // expert row 0..127
        const int c16 = id & 7;       // 16B column chunk within the KC*4B row
        const float* g = Wt + (size_t)row * (size_t)H + (size_t)(kc + c16 * 4);
        float*       l = dstBuf + row * BSTRIDE + c16 * 4;
#if HAVE_ASYNC_LDS
        __builtin_amdgcn_global_load_async_to_lds_b128((g_i4*)g, (l_i4*)l,
                                                       /*offset=*/0, /*cpol=*/0);
#else
        *(v4f*)l = *(const v4f*)g;    // sync fallback: global->VGPR->LDS
#endif
    }
}

// Grid: T/64 blocks, 128 threads (4 wave32 waves). Wave w computes logits for
// tokens [b*64 + w*16, +16) x 128 experts with v_wmma_f32_16x16x4_f32, B staged
// through double-buffered LDS (async), A register-prefetched one chunk ahead.
__global__ __launch_bounds__(128)
void gptoss_topk_router_kernel(const float* __restrict__ X,     // [T, H]
                               const float* __restrict__ Wt,    // [E, H]
                               const float* __restrict__ bias,  // [E]
                               float* __restrict__ outTop,      // [T, 4]
                               float* __restrict__ outIdx,      // [T, 4] (as float)
                               float* __restrict__ outLog,      // [T, E]
                               int H, int E)
{
    __shared__ float bs[2][128 * BSTRIDE];      // staged weights, double-buffered
    __shared__ float ls[4 * 16 * LDS_STRIDE];   // logits staging for top-k

    const int tid  = threadIdx.x;
    const int wave = tid >> 5;
    const int lane = tid & 31;
    const int half = lane >> 4;   // 0: K={0,1}, 1: K={2,3}
    const int nrow = lane & 15;   // A: token row; B/C: N (expert) index

    const int tokBase = blockIdx.x * 64 + wave * 16;
    const size_t aBase = (size_t)(tokBase + nrow) * (size_t)H + (size_t)(2 * half);
    const int nch = H / KC;       // 2880/32 = 90

    v8f acc[8] = {};
    v2f a_cur[8], a_nxt[8];

    // Prologue: stage chunk 0 weights + chunk 0 A tiles.
    stage_weights(Wt, H, 0, bs[0], tid);
    #pragma unroll
    for (int j = 0; j < 8; ++j)
        a_cur[j] = *(const v2f*)(X + aBase + (size_t)(4 * j));

    int buf = 0;
    for (int c = 0; c < nch; ++c) {
        const bool more = (c + 1 < nch);
        if (more) {
            stage_weights(Wt, H, (c + 1) * KC, bs[buf ^ 1], tid);
            const size_t an = aBase + (size_t)((c + 1) * KC);
            #pragma unroll
            for (int j = 0; j < 8; ++j)
                a_nxt[j] = *(const v2f*)(X + an + (size_t)(4 * j));
            wait_asynccnt_8();   // <=8 outstanding => chunk c fully landed
        } else {
            wait_asynccnt_0();
        }
        __syncthreads();         // chunk c visible to all waves; buf^1 free to fill

        // 8 k-steps x 8 N-tiles = 64 WMMAs on LDS-resident B.
        const float* bb = &bs[buf][nrow * BSTRIDE + 2 * half];
        #pragma unroll
        for (int j = 0; j < 8; ++j) {
            #pragma unroll
            for (int nt = 0; nt < 8; ++nt) {
                v2f b = *(const v2f*)(bb + nt * 16 * BSTRIDE + 4 * j);
                acc[nt] = __builtin_amdgcn_wmma_f32_16x16x4_f32(
                    false, a_cur[j], false, b, (short)0, acc[nt], false, false);
            }
        }
        __syncthreads();         // all waves done reading buf before it refills

        if (more) {
            #pragma unroll
            for (int j = 0; j < 8; ++j) a_cur[j] = a_nxt[j];
        }
        buf ^= 1;
    }

    // Bias add + store logits to global and LDS.
    // C layout: VGPR v, lane L -> M = v + 8*(L>>4), N = (L&15).
    #pragma unroll
    for (int nt = 0; nt < 8; ++nt) {
        const int   ex = nt * 16 + nrow;
        const float bv = bias[ex];
        #pragma unroll
        for (int v = 0; v < 8; ++v) {
            const int   tl  = v + 8 * half;
            const float val = acc[nt][v] + bv;
            outLog[(size_t)(tokBase + tl) * (size_t)E + (size_t)ex] = val;
            ls[(wave * 16 + tl) * LDS_STRIDE + ex] = val;
        }
    }

    __syncthreads();

    // Top-4 + softmax: lanes 0..15 each own one token of this wave's stripe.
    if (lane < 16) {
        const float* row = &ls[(wave * 16 + lane) * LDS_STRIDE];
        float v0 = -3.402823466e38f, v1 = v0, v2 = v0, v3 = v0;
        int   i0 = 0, i1 = 0, i2 = 0, i3 = 0;
        for (int e = 0; e < E; ++e) {
            const float v = row[e];
            // strict '>' => earliest index wins ties (matches jax.lax.top_k)
            if (v > v2) {
                if (v > v0)      { v3 = v2; i3 = i2; v2 = v1; i2 = i1; v1 = v0; i1 = i0; v0 = v; i0 = e; }
                else if (v > v1) { v3 = v2; i3 = i2; v2 = v1; i2 = i1; v1 = v;  i1 = e; }
                else             { v3 = v2; i3 = i2; v2 = v;  i2 = e; }
            } else if (v > v3)   { v3 = v;  i3 = e; }
        }

        const float m  = v0;
        const float e0 = __expf(v0 - m);
        const float e1 = __expf(v1 - m);
        const float e2 = __expf(v2 - m);
        const float e3 = __expf(v3 - m);
        const float inv = 1.0f / (e0 + e1 + e2 + e3);

        const size_t t4 = (size_t)(tokBase + lane) * 4;
        outTop[t4 + 0] = e0 * inv;
        outTop[t4 + 1] = e1 * inv;
        outTop[t4 + 2] = e2 * inv;
        outTop[t4 + 3] = e3 * inv;
        outIdx[t4 + 0] = (float)i0;
        outIdx[t4 + 1] = (float)i1;
        outIdx[t4 + 2] = (float)i2;
        outIdx[t4 + 3] = (float)i3;
    }
}

extern "C" void kernel_launch(void* const* d_in, const int* in_sizes, int n_in,
                              void* d_out, int out_size, void* d_ws, size_t ws_size,
                              hipStream_t stream) {
    const float* X    = (const float*)d_in[0];   // hidden_states [T, H]
    const float* Wt   = (const float*)d_in[1];   // weight [E, H]
    const float* bias = (const float*)d_in[2];   // bias [E]

    const int E = in_sizes[2];                   // 128
    const int H = in_sizes[1] / E;               // 2880
    const int T = in_sizes[0] / H;               // 16384

    float* out    = (float*)d_out;
    float* outTop = out;                          // [T,4]
    float* outIdx = out + (size_t)T * 4;          // [T,4] (as float)
    float* outLog = out + (size_t)T * 8;          // [T,E]

    dim3 block(128);
    dim3 grid(T / 64);
    gptoss_topk_router_kernel<<<grid, block, 0, stream>>>(
        X, Wt, bias, outTop, outIdx, outLog, H, E);
}